// LThuLoss_89696097010276
// MI455X (gfx1250) — compile-verified
//
#include <hip/hip_runtime.h>

// Native ext-vector types (needed for __builtin_nontemporal_load and WMMA operands)
typedef __attribute__((ext_vector_type(4))) float f4;
typedef __attribute__((ext_vector_type(2))) float v2f;
typedef __attribute__((ext_vector_type(8))) float v8f;

#define TILE 128           // KERNEL_SIZE
#define IMGW 1024          // image width/height
#define NBLK 2048          // 32 samples * 8 * 8 tiles
#define PIX_PER_TILE (TILE * TILE)   // 16384
#define BATCH 32

// ---------------------------------------------------------------------------
// Pass 1: one workgroup per 128x128 tile. Stream (fake - real) with 128-bit
// non-temporal loads and produce a single signed tile-sum into ws[blk].
// 256 threads = 8 waves of 32; each thread reads 16 float4 from each image.
// ---------------------------------------------------------------------------
__global__ __launch_bounds__(256) void lthu_block_sum(const float* __restrict__ fake,
                                                      const float* __restrict__ real,
                                                      float* __restrict__ ws) {
    __shared__ float wave_sum[8];

    const int tid  = threadIdx.x;
    const int blk  = blockIdx.x;         // 0..2047
    const int bi   = blk >> 6;           // sample (64 tiles per sample)
    const int tile = blk & 63;
    const int by   = tile >> 3;          // tile row 0..7
    const int bx   = tile & 7;           // tile col 0..7

    const int col4 = tid & 31;           // float4-column within tile (0..31)
    const int row0 = tid >> 5;           // 0..7

    const size_t base = (size_t)bi * (size_t)(IMGW * IMGW)
                      + (size_t)(by * TILE) * IMGW
                      + (size_t)(bx * TILE)
                      + (size_t)col4 * 4;

    float s = 0.0f;
#pragma unroll
    for (int r = 0; r < 16; ++r) {
        const size_t off = base + (size_t)(row0 + r * 8) * IMGW;
        f4 fv = __builtin_nontemporal_load((const f4*)(fake + off));
        f4 rv = __builtin_nontemporal_load((const f4*)(real + off));
        s += (fv.x - rv.x) + (fv.y - rv.y) + (fv.z - rv.z) + (fv.w - rv.w);
    }

    // wave32 butterfly reduction
#pragma unroll
    for (int o = 16; o > 0; o >>= 1) s += __shfl_xor(s, o, 32);
    if ((tid & 31) == 0) wave_sum[tid >> 5] = s;
    __syncthreads();

    if (tid == 0) {
        float t = 0.0f;
#pragma unroll
        for (int w = 0; w < 8; ++w) t += wave_sum[w];
        ws[blk] = t;   // signed tile sum (abs applied in pass 2)
    }
}

// ---------------------------------------------------------------------------
// Pass 2: single workgroup reduces 2048 |tile_sum| values to the scalar mean.
// The last 32->1 cross-lane reduction is done with V_WMMA_F32_16X16X4_F32:
//   A (16x4): A[m][0] = q(lane m), A[m][2] = q(lane 16+m), rest 0
//   B (4x16): all ones  ->  D[m][n] = q_m + q_{m+16}  (row-sum replicated)
// Each lane then sums its 8 D-VGPRs (rows m = r + 8*(lane>=16)) and one
// shfl_xor(16) combines the two halves into the full total.
// ---------------------------------------------------------------------------
__global__ __launch_bounds__(256) void lthu_final_reduce(const float* __restrict__ ws,
                                                         float* __restrict__ out) {
    __shared__ float part[256];
    const int tid = threadIdx.x;

    float p = 0.0f;
#pragma unroll
    for (int j = 0; j < 8; ++j) p += fabsf(ws[tid + 256 * j]);
    part[tid] = p;
    __syncthreads();

    if (tid < 32) {                      // whole wave 0: EXEC is all-ones here
        float q = 0.0f;
#pragma unroll
        for (int j = 0; j < 8; ++j) q += part[tid * 8 + j];

        v2f a; a.x = q;    a.y = 0.0f;   // A-matrix: K=0 (lanes 0-15) / K=2 (lanes 16-31)
        v2f b; b.x = 1.0f; b.y = 1.0f;   // B-matrix: all ones
        v8f c = {};                      // C = 0
        v8f d = __builtin_amdgcn_wmma_f32_16x16x4_f32(
            /*neg_a=*/false, a, /*neg_b=*/false, b,
            /*c_mod=*/(short)0, c, /*reuse_a=*/false, /*reuse_b=*/false);

        float sub = 0.0f;
#pragma unroll
        for (int r = 0; r < 8; ++r) sub += d[r];   // rows m = r + 8*(lane>=16)
        sub += __shfl_xor(sub, 16, 32);            // combine the two half-sums

        if (tid == 0)
            out[0] = sub * (1.0f / ((float)PIX_PER_TILE * (float)BATCH));
    }
}

// ---------------------------------------------------------------------------
extern "C" void kernel_launch(void* const* d_in, const int* in_sizes, int n_in,
                              void* d_out, int out_size, void* d_ws, size_t ws_size,
                              hipStream_t stream) {
    const float* fake = (const float*)d_in[0];
    const float* real = (const float*)d_in[1];
    float*       ws   = (float*)d_ws;    // 2048 floats = 8 KB scratch
    float*       out  = (float*)d_out;

    lthu_block_sum<<<NBLK, 256, 0, stream>>>(fake, real, ws);
    lthu_final_reduce<<<1, 256, 0, stream>>>(ws, out);
}